// PointLinkTrainer_77670188581643
// MI455X (gfx1250) — compile-verified
//
#include <hip/hip_runtime.h>
#include <hip/hip_bf16.h>

#define GRID 14
#define NC   20
// plane = one corner-index slice of one image: [14,14,4,51] floats
#define PLANE_ELEMS (GRID * GRID * 4 * 51)   // 39984
#define PLANE_VEC4  (PLANE_ELEMS / 4)        // 9996
#define NROWS       (GRID * GRID * 4)        // 784 rows of 51 channels

typedef __attribute__((ext_vector_type(2))) float v2f;
typedef __attribute__((ext_vector_type(8))) float v8f;
typedef __attribute__((ext_vector_type(4))) unsigned int u32x4;
typedef __attribute__((ext_vector_type(4))) int i32x4;
typedef __attribute__((ext_vector_type(8))) int i32x8;

#if __has_builtin(__builtin_amdgcn_tensor_load_to_lds) && \
    __has_builtin(__builtin_amdgcn_s_wait_tensorcnt)
#define HAS_TDM 1
#endif

// ---------------------------------------------------------------------------
// Main kernel: one block per (image, corner-plane k). blockIdx.x = img*4 + k.
// ---------------------------------------------------------------------------
__global__ __launch_bounds__(256)
void pointlink_loss_kernel(const float* __restrict__ preds,
                           const float* __restrict__ bboxes,
                           const int*   __restrict__ labels,
                           const int*   __restrict__ Wptr,
                           float*       __restrict__ partials) {
  const int img = blockIdx.x >> 2;
  const int k   = blockIdx.x & 3;
  const int tid = threadIdx.x;

  // ---- per-image prep data in LDS --------------------------------------
  __shared__ __align__(16) float s_bbox[128]; // staged bbox tile (TDM dest)
  __shared__ float s_pdx[128], s_pdy[128];    // corner fractional offsets
  __shared__ int   s_pcx[128], s_pcy[128];    // corner cell coords
  __shared__ float s_cdx[32],  s_cdy[32];     // center fractional offsets
  __shared__ int   s_ccx[32],  s_ccy[32];     // center cell coords
  __shared__ int   s_label[32];
  __shared__ int   s_assign[GRID * GRID];     // point index per cell (min wins)
  __shared__ float4 s_rec[GRID * GRID * 2];   // per-(cell, j-half) target rec
  __shared__ float s_wm[GRID * GRID * 51];    // matched-masked channel weight
  __shared__ float s_wave[16];                // per-wave-half partial sums

  for (int i = tid; i < GRID * GRID; i += 256) s_assign[i] = 0x7FFFFFFF;

  const float Wf    = (float)(*Wptr);
  const float scale = (float)(GRID - 1) / Wf;

  if (tid < 32) {
#ifdef HAS_TDM
    // --- Tensor Data Mover: DMA the 32x4 f32 bbox tile into LDS ---------
    // D# group0: count=1 | lds_addr | global_addr(57b) | type=2
    // D# group1: data_size=4B, tensor_dim0=tile_dim0=128, dim1=1, stride=128
    {
      const unsigned long long ga =
          (unsigned long long)(const void*)(bboxes + (long long)img * 128);
      const unsigned lds_off =
          (unsigned)(unsigned long long)(const void*)&s_bbox[0];
      u32x4 g0;
      g0[0] = 1u;                       // count=1 (valid user descriptor)
      g0[1] = lds_off;                  // lds_addr
      g0[2] = (unsigned)ga;             // global_addr[31:0]
      g0[3] = (unsigned)((ga >> 32) & 0x1FFFFFFu) | 0x80000000u; // [56:32]|type2
      i32x8 g1;
      g1[0] = 2 << 16;                  // data_size = 4 bytes
      g1[1] = 128 << 16;                // tensor_dim0 = 128 (low 16 bits)
      g1[2] = 1 << 16;                  // tensor_dim1 = 1
      g1[3] = 128 << 16;                // tile_dim0 = 128
      g1[4] = 1;                        // tile_dim1 = 1
      g1[5] = 128;                      // tensor_dim0_stride = 128
      g1[6] = 0;
      g1[7] = 0;
      i32x4 g2; g2[0] = 0; g2[1] = 0; g2[2] = 0; g2[3] = 0;
      i32x4 g3; g3[0] = 0; g3[1] = 0; g3[2] = 0; g3[3] = 0;
#if defined(__clang_major__) && __clang_major__ >= 23
      i32x8 g4; g4[0]=0; g4[1]=0; g4[2]=0; g4[3]=0; g4[4]=0; g4[5]=0; g4[6]=0; g4[7]=0;
      __builtin_amdgcn_tensor_load_to_lds(g0, g1, g2, g3, g4, 0);
#else
      __builtin_amdgcn_tensor_load_to_lds(g0, g1, g2, g3, 0);
#endif
      __builtin_amdgcn_s_wait_tensorcnt(0);   // issuing wave consumes it
    }
    const float4 bb = ((const float4*)s_bbox)[tid];
#else
    const float4 bb = ((const float4*)bboxes)[img * 32 + tid];
#endif
    const float b0 = bb.x * scale, b1 = bb.y * scale;
    const float b2 = bb.z * scale, b3 = bb.w * scale;
    const float pxs[4] = {b0, b2, b0, b2};
    const float pys[4] = {b1, b1, b3, b3};
#pragma unroll
    for (int c4 = 0; c4 < 4; ++c4) {
      const int p   = tid * 4 + c4;
      const int icx = (int)floorf(pxs[c4]);
      const int icy = (int)floorf(pys[c4]);
      s_pcx[p] = icx;
      s_pcy[p] = icy;
      s_pdx[p] = pxs[c4] - (float)icx;
      s_pdy[p] = pys[c4] - (float)icy;
    }
    const float cx = (b0 + b2) * 0.5f, cy = (b1 + b3) * 0.5f;
    const int iccx = (int)floorf(cx), iccy = (int)floorf(cy);
    s_ccx[tid] = iccx;  s_ccy[tid] = iccy;
    s_cdx[tid] = cx - (float)iccx;
    s_cdy[tid] = cy - (float)iccy;
    s_label[tid] = labels[img * 32 + tid];
  }
  __syncthreads();

  // scatter: lowest point index wins per cell (matches reversed .at[].set)
  if (tid < 128) {
    const int cx = min(max(s_pcx[tid], 0), GRID - 1);
    const int cy = min(max(s_pcy[tid], 0), GRID - 1);
    atomicMin(&s_assign[cx * GRID + cy], tid);
  }
  __syncthreads();

  // ---- build matched-masked per-(cell,channel) weight table ------------
  for (int i = tid; i < GRID * GRID * 51; i += 256) {
    const int cell = i / 51;
    const int c    = i - cell * 51;
    float w = 1.0f / (float)GRID;                  // link channels 23..50
    if (c == 0)      w = 1.0f;                     // confidence
    else if (c < 21) w = 1.0f / (float)NC;         // class
    else if (c < 23) w = 0.5f;                     // coords
    s_wm[i] = (s_assign[cell] != 0x7FFFFFFF) ? w : 0.0f;
  }

  // ---- build packed per-(cell, half) target records --------------------
  // packed bits: [5:0] conf idx, [11:6] label+1, [17:12] linkx+23,
  //              [23:18] linky+37, [24] matched
  if (tid < GRID * GRID) {
    const int  a   = s_assign[tid];
    const bool m   = (a != 0x7FFFFFFF);
    const int  pt  = m ? a : 0;
    const int  box = pt >> 2;
    const int c0i = m ? 0 : 63;
    const int l1  = m ? (s_label[box] + 1) : 63;
    const int mb  = m ? (1 << 24) : 0;
    // half 0 (j<2): coords = corner offsets of pt, links = center cell
    const int gx0 = m ? (s_ccx[box] + 23) : 63;
    const int gy0 = m ? (s_ccy[box] + 37) : 63;
    float4 r0;
    r0.x = m ? s_pdx[pt] : 0.0f;
    r0.y = m ? s_pdy[pt] : 0.0f;
    r0.z = __int_as_float(c0i | (l1 << 6) | (gx0 << 12) | (gy0 << 18) | mb);
    r0.w = 0.0f;
    s_rec[tid * 2 + 0] = r0;
    // half 1 (j>=2): coords = center offsets, links = corner-k cell of box
    const int gx1 = m ? (s_pcx[(box << 2) | k] + 23) : 63;
    const int gy1 = m ? (s_pcy[(box << 2) | k] + 37) : 63;
    float4 r1;
    r1.x = m ? s_cdx[box] : 0.0f;
    r1.y = m ? s_cdy[box] : 0.0f;
    r1.z = __int_as_float(c0i | (l1 << 6) | (gx1 << 12) | (gy1 << 18) | mb);
    r1.w = 0.0f;
    s_rec[tid * 2 + 1] = r1;
  }
  __syncthreads();

  const float* plane = preds + (long long)(img * 4 + k) * PLANE_ELEMS;
  float acc = 0.0f;

  // ---- correction pass: cross/constant terms, <=6 gathers per row ------
  // Full loss = bulk Sum(w*v^2)  +  per-row corrections below.
  for (int row = tid; row < NROWS; row += 256) {
    const float4 rc = s_rec[row >> 1];
    const int    pk = __float_as_int(rc.z);
    const float* rp = plane + row * 51;
    if ((pk >> 24) != 0) {                           // matched row
      const int l1 = (pk >> 6) & 63;
      const int gx = (pk >> 12) & 63;
      const int gy = (pk >> 18) & 63;
      const float v0  = rp[0],  vl  = rp[l1];
      const float v21 = rp[21], v22 = rp[22];
      const float vgx = rp[gx], vgy = rp[gy];
      acc += (1.0f - 2.0f * v0)                                   // conf
           + (1.0f / (float)NC) * (1.0f - 2.0f * vl)              // class
           + 0.5f * (rc.x * rc.x + rc.y * rc.y)
           - (v21 * rc.x + v22 * rc.y)                            // coords
           + (1.0f / (float)GRID) * (2.0f - 2.0f * (vgx + vgy));  // links
    } else {
      const float v0 = rp[0];
      acc += v0 * v0;        // unmatched conf (bulk contributed 0 via wm=0)
    }
  }

  // ---- bulk pass: acc += wm[cell*51+c] * v^2, streaming b128 -----------
  const float4* p4 = (const float4*)plane;
  const int idx0 = tid * 4;
  int cu       = idx0 % 51;                 // channel 0..50
  const int r0 = idx0 / 51;
  int jj       = r0 & 3;                    // j within cell
  int cellBase = (r0 >> 2) * 51;            // cell * 51

  for (int i4 = tid; i4 < PLANE_VEC4; i4 += 256) {
    const float4 v = p4[i4];
    if (i4 + 2048 < PLANE_VEC4)
      __builtin_prefetch(&p4[i4 + 2048], 0, 0);   // global_prefetch_b8
    const float vv[4] = {v.x, v.y, v.z, v.w};
#pragma unroll
    for (int u = 0; u < 4; ++u) {
      const float w = s_wm[cellBase + cu];
      acc = fmaf(w * vv[u], vv[u], acc);
      // advance one element with channel / row wrap (branchless selects)
      cu += 1;
      const bool wrap = (cu == 51);
      cu = wrap ? 0 : cu;
      jj += wrap ? 1 : 0;
      const bool jw = (jj == 4);
      jj = jw ? 0 : jj;
      cellBase += jw ? 51 : 0;
    }
    cellBase += 255;   // remaining stride: 1020 elements = 20 rows = 5 cells
  }

  // ---- exact wave32 reduction via V_WMMA_F32_16X16X4_F32 ---------------
  // A[16x4]: lanes 0-15 -> {K0=acc, K1=0}, lanes 16-31 -> {K2=acc, K3=0}
  // B[4x16] = all ones  =>  D[m][n] = acc[m] + acc[m+16]  (exact f32 adds)
  v2f a2;  a2.x = acc;  a2.y = 0.0f;
  v2f b2;  b2.x = 1.0f; b2.y = 1.0f;
  v8f c8 = {0.f, 0.f, 0.f, 0.f, 0.f, 0.f, 0.f, 0.f};
  v8f d8 = __builtin_amdgcn_wmma_f32_16x16x4_f32(
      false, a2, false, b2, (short)0, c8, false, false);
  float s8 = d8[0] + d8[1] + d8[2] + d8[3] + d8[4] + d8[5] + d8[6] + d8[7];

  // lane 0 holds sum_{m=0..7} D[m][0], lane 16 holds sum_{m=8..15} D[m][0]
  const int lane = tid & 31;
  const int wave = tid >> 5;                 // 8 waves per block
  if (lane == 0)  s_wave[wave * 2 + 0] = s8;
  if (lane == 16) s_wave[wave * 2 + 1] = s8;
  __syncthreads();

  if (tid == 0) {                            // deterministic fixed-order sum
    float bsum = 0.0f;
#pragma unroll
    for (int i = 0; i < 16; ++i) bsum += s_wave[i];
    partials[blockIdx.x] = bsum;
  }
}

// ---------------------------------------------------------------------------
// Deterministic final reduction of per-block partials.
// ---------------------------------------------------------------------------
__global__ __launch_bounds__(256)
void reduce_partials_kernel(const float* __restrict__ ws,
                            float* __restrict__ out, int n) {
  __shared__ float s[256];
  float a = 0.0f;
  for (int i = threadIdx.x; i < n; i += 256) a += ws[i];
  s[threadIdx.x] = a;
  __syncthreads();
  for (int off = 128; off > 0; off >>= 1) {
    if (threadIdx.x < off) s[threadIdx.x] += s[threadIdx.x + off];
    __syncthreads();
  }
  if (threadIdx.x == 0) out[0] = s[0];
}

extern "C" void kernel_launch(void* const* d_in, const int* in_sizes, int n_in,
                              void* d_out, int out_size, void* d_ws, size_t ws_size,
                              hipStream_t stream) {
  const float* preds  = (const float*)d_in[0];
  const float* bboxes = (const float*)d_in[1];
  const int*   labels = (const int*)d_in[2];
  const int*   Wptr   = (const int*)d_in[3];
  float*       out    = (float*)d_out;
  float*       parts  = (float*)d_ws;

  const int batch   = in_sizes[0] / (4 * PLANE_ELEMS);  // 512
  const int nblocks = batch * 4;

  pointlink_loss_kernel<<<nblocks, 256, 0, stream>>>(preds, bboxes, labels,
                                                     Wptr, parts);
  reduce_partials_kernel<<<1, 256, 0, stream>>>(parts, out, nblocks);
}